// CustomRNN_11862699671823
// MI455X (gfx1250) — compile-verified
//
#include <hip/hip_runtime.h>
#include <hip/hip_bf16.h>
#include <math.h>

typedef float v2f __attribute__((ext_vector_type(2)));
typedef float v4f __attribute__((ext_vector_type(4)));
typedef float v8f __attribute__((ext_vector_type(8)));
typedef int   v4i __attribute__((ext_vector_type(4)));

#define BATCH   64
#define TSEQ    2048
#define FEAT    128
#define UNITS   256

// LDS padding strides (floats): even (8B align for ds b64) and bank-spreading.
#define WT_STRIDE 132   // phase-1 W^T rows (len FEAT)
#define UT_STRIDE 258   // phase-2 U^T rows (len UNITS)
#define H_STRIDE  260   // phase-2 h rows (len UNITS)
#define XW_STRIDE 264   // phase-2 xw staging rows (len UNITS); 264*4 % 16 == 0

#define GLOBAL_AS __attribute__((address_space(1)))
#define LDS_AS    __attribute__((address_space(3)))

#if defined(__gfx1250__) && __has_builtin(__builtin_amdgcn_global_load_async_to_lds_b128)
#define HAVE_ASYNC_LDS 1
#else
#define HAVE_ASYNC_LDS 0
#endif

__device__ __forceinline__ float fast_tanh(float x) {
#if defined(__gfx1250__) && __has_builtin(__builtin_amdgcn_tanhf)
  return __builtin_amdgcn_tanhf(x);     // native V_TANH_F32 (TRANS pipe)
#else
  return tanhf(x);
#endif
}

__device__ __forceinline__ void wait_async_lds() {
#if HAVE_ASYNC_LDS
#if __has_builtin(__builtin_amdgcn_s_wait_asynccnt)
  __builtin_amdgcn_s_wait_asynccnt(0);
#else
  asm volatile("s_wait_asynccnt 0x0" ::: "memory");
#endif
#endif
}

// Stage one 16x256-f32 xw tile (16 KB) for timestep t into LDS.
// 1024 b128 chunks across 512 threads (2 each). Async on CDNA5, sync fallback.
__device__ __forceinline__ void stage_xw(const float* __restrict__ XW,
                                         float* __restrict__ dst,
                                         int brow0, int t, int tid) {
#pragma unroll
  for (int s = 0; s < 2; ++s) {
    const int c   = tid + 512 * s;        // 0..1023
    const int row = c >> 6;               // 64 b128 chunks per row
    const int col = (c & 63) << 2;        // float offset, 16B aligned
    const float* g = XW + ((size_t)(brow0 + row) * TSEQ + t) * UNITS + col;
    float* l = dst + row * XW_STRIDE + col;
#if HAVE_ASYNC_LDS
    __builtin_amdgcn_global_load_async_to_lds_b128(
        (GLOBAL_AS v4i*)(GLOBAL_AS void*)g,
        (LDS_AS v4i*)(LDS_AS void*)l, 0, 0);
#else
    *(v4f*)l = *(const v4f*)g;
#endif
  }
}

// ---------------------------------------------------------------------------
// Phase 1: XW[m, n] = X[m, :] @ W[:, n] + b[n]   (M=131072, K=128, N=256)
// Memory-bound (~200 MB traffic). 16 waves/block, one N-tile per wave,
// W^T staged in LDS once per block, 32 M-tiles per block.
// ---------------------------------------------------------------------------
__global__ void __launch_bounds__(512)
rnn_xw_gemm(const float* __restrict__ X, const float* __restrict__ W,
            const float* __restrict__ bias, float* __restrict__ XW,
            int mtiles_per_block) {
  extern __shared__ float lds[];            // Wt: [UNITS][WT_STRIDE]
  float* Wt = lds;
  const int tid = threadIdx.x;

  for (int i = tid; i < FEAT * UNITS; i += 512) {   // coalesced read, transpose
    const int k = i / UNITS, n = i % UNITS;
    Wt[n * WT_STRIDE + k] = W[i];
  }
  __syncthreads();

  const int wave = tid >> 5;
  const int lane = tid & 31;
  const int half = lane >> 4;
  const int r    = lane & 15;
  const int n0   = wave * 16;
  const float bn = bias[n0 + r];

  for (int it = 0; it < mtiles_per_block; ++it) {
    const int row0 = (blockIdx.x * mtiles_per_block + it) * 16;
    const float* arow = X + (size_t)(row0 + r) * FEAT;

    v8f acc0, acc1;                         // dual accumulators: 2x WMMA ILP
#pragma unroll
    for (int j = 0; j < 8; ++j) { acc0[j] = bn; acc1[j] = 0.0f; }

#pragma unroll 4
    for (int k0 = 0; k0 < FEAT; k0 += 8) {
      v2f a0 = *(const v2f*)(arow + k0 + 2 * half);
      v2f b0 = *(const v2f*)(&Wt[(n0 + r) * WT_STRIDE + k0 + 2 * half]);
      acc0 = __builtin_amdgcn_wmma_f32_16x16x4_f32(false, a0, false, b0,
                                                   (short)0, acc0, false, false);
      v2f a1 = *(const v2f*)(arow + k0 + 4 + 2 * half);
      v2f b1 = *(const v2f*)(&Wt[(n0 + r) * WT_STRIDE + k0 + 4 + 2 * half]);
      acc1 = __builtin_amdgcn_wmma_f32_16x16x4_f32(false, a1, false, b1,
                                                   (short)0, acc1, false, false);
    }

    float* orow = XW + (size_t)row0 * UNITS + n0 + r;
#pragma unroll
    for (int j = 0; j < 8; ++j)             // keep XW resident in L2 (RT stores)
      orow[(size_t)(j + 8 * half) * UNITS] = acc0[j] + acc1[j];
  }
}

// ---------------------------------------------------------------------------
// Phase 2: h_t = tanh(xw_t + h_{t-1} @ U). Batch rows are independent ->
// 4 blocks x 16 rows, zero inter-block traffic. U^T resident in LDS (~258 KB)
// for all 2048 steps; next xw tile async-prefetched into LDS each step.
// ---------------------------------------------------------------------------
__global__ void __launch_bounds__(512)
rnn_recurrence(const float* __restrict__ XW, const float* __restrict__ U,
               float* __restrict__ out) {
  extern __shared__ float lds[];
  float* Ut    = lds;                              // [UNITS][UT_STRIDE]
  float* hbuf  = Ut + UNITS * UT_STRIDE;           // [16][H_STRIDE]
  float* xwbuf = hbuf + 16 * H_STRIDE;             // 2 x [16][XW_STRIDE]
  const int tid = threadIdx.x;

  const int wave  = tid >> 5;
  const int lane  = tid & 31;
  const int half  = lane >> 4;
  const int r     = lane & 15;
  const int n0    = wave * 16;
  const int brow0 = blockIdx.x * 16;

  for (int i = tid; i < UNITS * UNITS; i += 512) { // U^T into LDS (coalesced)
    const int k = i / UNITS, n = i % UNITS;
    Ut[n * UT_STRIDE + k] = U[i];
  }
  for (int i = tid; i < 16 * H_STRIDE; i += 512) hbuf[i] = 0.0f;   // h0 = 0
  stage_xw(XW, xwbuf, brow0, 0, tid);              // xw tile for t=0
  wait_async_lds();
  __syncthreads();

  for (int t = 0; t < TSEQ; ++t) {
    const int p = t & 1;
    if (t + 1 < TSEQ)                              // overlap next fetch w/ WMMAs
      stage_xw(XW, xwbuf + (p ^ 1) * (16 * XW_STRIDE), brow0, t + 1, tid);

    const float* xwt = xwbuf + p * (16 * XW_STRIDE);
    v8f acc0, acc1;
#pragma unroll
    for (int j = 0; j < 8; ++j) {
      acc0[j] = xwt[(j + 8 * half) * XW_STRIDE + n0 + r];
      acc1[j] = 0.0f;
    }

#pragma unroll 4
    for (int k0 = 0; k0 < UNITS; k0 += 8) {        // two independent WMMA chains
      v2f a0 = *(const v2f*)(hbuf + r * H_STRIDE + k0 + 2 * half);
      v2f b0 = *(const v2f*)(&Ut[(n0 + r) * UT_STRIDE + k0 + 2 * half]);
      acc0 = __builtin_amdgcn_wmma_f32_16x16x4_f32(false, a0, false, b0,
                                                   (short)0, acc0, false, false);
      v2f a1 = *(const v2f*)(hbuf + r * H_STRIDE + k0 + 4 + 2 * half);
      v2f b1 = *(const v2f*)(&Ut[(n0 + r) * UT_STRIDE + k0 + 4 + 2 * half]);
      acc1 = __builtin_amdgcn_wmma_f32_16x16x4_f32(false, a1, false, b1,
                                                   (short)0, acc1, false, false);
    }

#pragma unroll
    for (int j = 0; j < 8; ++j) acc0[j] = fast_tanh(acc0[j] + acc1[j]);

    __syncthreads();                               // all h_{t-1} reads done

    float* obase = out + ((size_t)brow0 * TSEQ + t) * UNITS + n0 + r;
#pragma unroll
    for (int j = 0; j < 8; ++j) {
      __builtin_nontemporal_store(acc0[j],
          obase + (size_t)(j + 8 * half) * TSEQ * UNITS);   // write-once output
      hbuf[(j + 8 * half) * H_STRIDE + n0 + r] = acc0[j];
    }

    wait_async_lds();                              // own async tile complete
    __syncthreads();                               // h_t + xw_{t+1} visible
  }
}

// ---------------------------------------------------------------------------
extern "C" void kernel_launch(void* const* d_in, const int* in_sizes, int n_in,
                              void* d_out, int out_size, void* d_ws, size_t ws_size,
                              hipStream_t stream) {
  const float* X  = (const float*)d_in[0];   // (64, 2048, 128)
  const float* W  = (const float*)d_in[1];   // (128, 256)
  const float* U  = (const float*)d_in[2];   // (256, 256)
  const float* b  = (const float*)d_in[3];   // (256,)
  float* out      = (float*)d_out;           // (64, 2048, 256)
  float* XW       = (float*)d_ws;            // (64*2048, 256) scratch, 128 MiB

  const int mtiles         = (BATCH * TSEQ) / 16;   // 8192
  const int blocks1        = 256;
  const int mtiles_per_blk = mtiles / blocks1;      // 32

  const size_t lds1 = (size_t)UNITS * WT_STRIDE * sizeof(float);   // ~132 KB
  hipLaunchKernelGGL(rnn_xw_gemm, dim3(blocks1), dim3(512), lds1, stream,
                     X, W, b, XW, mtiles_per_blk);

  const size_t lds2 = ((size_t)UNITS * UT_STRIDE + 16u * H_STRIDE +
                       2u * 16u * XW_STRIDE) * sizeof(float);      // ~307 KB
  hipLaunchKernelGGL(rnn_recurrence, dim3(BATCH / 16), dim3(512), lds2, stream,
                     XW, U, out);
}